// BlockV1_2413771620882
// MI455X (gfx1250) — compile-verified
//
#include <hip/hip_runtime.h>

// Problem constants (match reference)
#define B_      16
#define L_      32768
#define H_      32
#define N_      4
#define COND_   2

// Chunked-scan config
#define T_              128                 // chunk length (multiple of 16)
#define CHUNKS_         (L_ / T_)           // 256 chunks per sequence
#define WAVES_PER_BLOCK 8
#define BLOCK_THREADS   (WAVES_PER_BLOCK * 32)

typedef float v2f __attribute__((ext_vector_type(2)));
typedef float v8f __attribute__((ext_vector_type(8)));

// Raw hardware 2^x. Arg is always <= 0 at our call sites, so no range
// reduction is needed (deep-negative -> 0 is the correct limit; NaN
// propagates natively). The v_nop inside the asm satisfies the TRANS-op
// result hazard (1 instruction gap) without relying on the scheduler.
__device__ __forceinline__ float raw_exp2(float x) {
#if __has_builtin(__builtin_amdgcn_exp2f)
    return __builtin_amdgcn_exp2f(x);
#else
    float r;
    asm("v_exp_f32 %0, %1\n\tv_nop" : "=v"(r) : "v"(x));
    return r;
#endif
}

// Branch-free tanh: tanh(x) = sign(x) * (1 - e) / (1 + e), e = exp(-2|x|).
// ~6 straight-line VALU ops, no EXEC-mask branches, no cmp/cndmask ladders.
__device__ __forceinline__ float fast_tanh(float v) {
    float a = __builtin_fabsf(v);
    float e = raw_exp2(-2.8853900817779268f * a);   // exp(-2|v|)
    float t = (1.0f - e) * __builtin_amdgcn_rcpf(1.0f + e);
    return __builtin_copysignf(t, v);
}

// Per-channel decay lambda_n = exp(dt * A_n), A_n = -exp(log_A_real) + i*A_imag
// (one-time per-lane setup: keep precise libm here)
__device__ __forceinline__ void channel_decay(const float* __restrict__ log_dt,
                                              const float* __restrict__ log_A_real,
                                              const float* __restrict__ A_imag,
                                              int h, float* lam_re, float* lam_im,
                                              float* Are_out, float* Aim_out, float* dt_out) {
    float dt = expf(log_dt[h]);
    *dt_out = dt;
#pragma unroll
    for (int n = 0; n < N_; ++n) {
        float Are = -expf(log_A_real[h * N_ + n]);
        float Aim = A_imag[h * N_ + n];
        float er = expf(Are * dt);
        float s, c;
        sincosf(Aim * dt, &s, &c);
        lam_re[n] = er * c;
        lam_im[n] = er * s;
        Are_out[n] = Are;
        Aim_out[n] = Aim;
    }
}

// ---------------------------------------------------------------------------
// Phase 1: fused (x @ W^T -> tanh) projection via WMMA + per-chunk local scan.
// One wave per (batch, chunk); lane = channel h. Writes chunk end-state.
// carry layout: [(b*CHUNKS_+c)*H_ + h] * 8 floats (re,im interleaved, n=0..3)
// ---------------------------------------------------------------------------
__global__ __launch_bounds__(BLOCK_THREADS)
void s4d_phase1(const float* __restrict__ x, const float* __restrict__ lin_w,
                const float* __restrict__ lin_b,
                const float* __restrict__ log_dt, const float* __restrict__ log_A_real,
                const float* __restrict__ A_imag,
                float* __restrict__ carry) {
    __shared__ float lds[WAVES_PER_BLOCK][16][H_];

    const int lane = threadIdx.x & 31;
    const int wv   = threadIdx.x >> 5;
    const int w    = blockIdx.x * WAVES_PER_BLOCK + wv;   // global wave = (b, c)
    const int b    = w / CHUNKS_;
    const int c    = w % CHUNKS_;

    const int m16   = lane & 15;          // row-in-tile / col-in-tile
    const int khalf = (lane >> 4) << 1;   // K sub-pair: 0 or 2

    // B fragments: B[k][j] = W[j][k] (j = output channel), plus bias
    v2f bfrag[2][8];
    float bias[2];
#pragma unroll
    for (int jt = 0; jt < 2; ++jt) {
        int j = jt * 16 + m16;
        bias[jt] = lin_b[j];
#pragma unroll
        for (int kk = 0; kk < 8; ++kk) {
            int k = kk * 4 + khalf;
            bfrag[jt][kk] = *(const v2f*)&lin_w[j * H_ + k];
        }
    }

    float lam_re[N_], lam_im[N_], Are[N_], Aim[N_], dt;
    channel_decay(log_dt, log_A_real, A_imag, lane, lam_re, lam_im, Are, Aim, &dt);

    float sre[N_] = {0.f, 0.f, 0.f, 0.f};
    float sim[N_] = {0.f, 0.f, 0.f, 0.f};

    const long rbase0 = (long)b * L_ + (long)c * T_;

    for (int g = 0; g < T_ / 16; ++g) {
        // A fragment: 16 rows of x, full K=32 (loaded once, reused for both tiles)
        const float* xr = x + (rbase0 + g * 16 + m16) * H_;
        v2f afrag[8];
#pragma unroll
        for (int kk = 0; kk < 8; ++kk)
            afrag[kk] = *(const v2f*)&xr[kk * 4 + khalf];

#pragma unroll
        for (int jt = 0; jt < 2; ++jt) {
            v8f acc = {0.f, 0.f, 0.f, 0.f, 0.f, 0.f, 0.f, 0.f};
#pragma unroll
            for (int kk = 0; kk < 8; ++kk)
                acc = __builtin_amdgcn_wmma_f32_16x16x4_f32(
                    false, afrag[kk], false, bfrag[jt][kk], (short)0, acc, false, false);
#pragma unroll
            for (int v = 0; v < 8; ++v) {
                int m = ((lane >> 4) << 3) + v;   // row in 16-row group
                lds[wv][m][jt * 16 + m16] = fast_tanh(acc[v] + bias[jt]);
            }
        }
        // Same-wave DS ordering is guaranteed by HW (DS in-order per wave);
        // fence stops the compiler from reordering the loads above the stores.
        __asm__ volatile("" ::: "memory");

#pragma unroll 4
        for (int t = 0; t < 16; ++t) {
            float u = lds[wv][t][lane];
#pragma unroll
            for (int n = 0; n < N_; ++n) {
                float r = fmaf(lam_re[n], sre[n], fmaf(-lam_im[n], sim[n], u));
                float i = fmaf(lam_re[n], sim[n], lam_im[n] * sre[n]);
                sre[n] = r; sim[n] = i;
            }
        }
        __asm__ volatile("" ::: "memory");
    }

    float* outp = carry + ((long)(b * CHUNKS_ + c) * H_ + lane) * 8;
#pragma unroll
    for (int n = 0; n < N_; ++n) { outp[2 * n] = sre[n]; outp[2 * n + 1] = sim[n]; }
}

// ---------------------------------------------------------------------------
// Phase 2: in-place exclusive scan across chunks. 512 threads = (b, h) pairs.
// Combine operator: s_next = lambda^T * s + local.
// ---------------------------------------------------------------------------
__global__ __launch_bounds__(B_ * H_)
void s4d_phase2(const float* __restrict__ log_dt, const float* __restrict__ log_A_real,
                const float* __restrict__ A_imag,
                float* __restrict__ carry) {
    const int tid = threadIdx.x;
    const int b = tid >> 5, h = tid & 31;

    float dt = expf(log_dt[h]);
    float lTre[N_], lTim[N_];
#pragma unroll
    for (int n = 0; n < N_; ++n) {
        float Are = -expf(log_A_real[h * N_ + n]);
        float Aim = A_imag[h * N_ + n];
        float er = expf(Are * dt * (float)T_);
        float s, c;
        sincosf(Aim * dt * (float)T_, &s, &c);
        lTre[n] = er * c;
        lTim[n] = er * s;
    }

    float rre[N_] = {0.f, 0.f, 0.f, 0.f};
    float rim[N_] = {0.f, 0.f, 0.f, 0.f};
    for (int c = 0; c < CHUNKS_; ++c) {
        float* slot = carry + ((long)(b * CHUNKS_ + c) * H_ + h) * 8;
#pragma unroll
        for (int n = 0; n < N_; ++n) {
            float tre = slot[2 * n], tim = slot[2 * n + 1];
            slot[2 * n] = rre[n]; slot[2 * n + 1] = rim[n];   // state ENTERING chunk c
            float nr = fmaf(lTre[n], rre[n], fmaf(-lTim[n], rim[n], tre));
            float ni = fmaf(lTre[n], rim[n], fmaf(lTim[n], rre[n], tim));
            rre[n] = nr; rim[n] = ni;
        }
    }
}

// ---------------------------------------------------------------------------
// Phase 3: re-run fused projection + scan seeded with entry state; emit
// tanh(g * (2*Re(Ct . s) + D*u) + beta) with coalesced stores (lane = h).
// ---------------------------------------------------------------------------
__global__ __launch_bounds__(BLOCK_THREADS)
void s4d_phase3(const float* __restrict__ x, const float* __restrict__ lin_w,
                const float* __restrict__ lin_b,
                const float* __restrict__ log_dt, const float* __restrict__ log_A_real,
                const float* __restrict__ A_imag,
                const float* __restrict__ C_re, const float* __restrict__ C_im,
                const float* __restrict__ Dvec,
                const float* __restrict__ cond, const float* __restrict__ film_w,
                const float* __restrict__ film_b,
                const float* __restrict__ carry,
                float* __restrict__ out) {
    __shared__ float lds[WAVES_PER_BLOCK][16][H_];

    const int lane = threadIdx.x & 31;
    const int wv   = threadIdx.x >> 5;
    const int w    = blockIdx.x * WAVES_PER_BLOCK + wv;
    const int b    = w / CHUNKS_;
    const int c    = w % CHUNKS_;

    const int m16   = lane & 15;
    const int khalf = (lane >> 4) << 1;

    v2f bfrag[2][8];
    float bias[2];
#pragma unroll
    for (int jt = 0; jt < 2; ++jt) {
        int j = jt * 16 + m16;
        bias[jt] = lin_b[j];
#pragma unroll
        for (int kk = 0; kk < 8; ++kk)
            bfrag[jt][kk] = *(const v2f*)&lin_w[j * H_ + kk * 4 + khalf];
    }

    float lam_re[N_], lam_im[N_], Are[N_], Aim[N_], dt;
    channel_decay(log_dt, log_A_real, A_imag, lane, lam_re, lam_im, Are, Aim, &dt);

    // Ct = 2 * (C_re + i C_im) * (lambda - 1) / A    (factor 2 folded in)
    float Ctre[N_], Ctim[N_];
#pragma unroll
    for (int n = 0; n < N_; ++n) {
        float nre = lam_re[n] - 1.f, nim = lam_im[n];
        float den = Are[n] * Are[n] + Aim[n] * Aim[n];
        float qre = (nre * Are[n] + nim * Aim[n]) / den;   // (lambda-1)/A
        float qim = (nim * Are[n] - nre * Aim[n]) / den;
        float cr = C_re[lane * N_ + n], ci = C_im[lane * N_ + n];
        Ctre[n] = 2.f * (cr * qre - ci * qim);
        Ctim[n] = 2.f * (cr * qim + ci * qre);
    }

    // FiLM gain/bias for this (b, h): gb = cond @ film_w^T + film_b
    const float c0 = cond[b * COND_ + 0], c1 = cond[b * COND_ + 1];
    const float gg = fmaf(c0, film_w[lane * COND_ + 0],
                      fmaf(c1, film_w[lane * COND_ + 1], film_b[lane]));
    const float bb = fmaf(c0, film_w[(H_ + lane) * COND_ + 0],
                      fmaf(c1, film_w[(H_ + lane) * COND_ + 1], film_b[H_ + lane]));
    const float Dh = Dvec[lane];

    // Entry state for this chunk
    const float* init = carry + ((long)(b * CHUNKS_ + c) * H_ + lane) * 8;
    float sre[N_], sim[N_];
#pragma unroll
    for (int n = 0; n < N_; ++n) { sre[n] = init[2 * n]; sim[n] = init[2 * n + 1]; }

    const long rbase0 = (long)b * L_ + (long)c * T_;

    for (int g = 0; g < T_ / 16; ++g) {
        const float* xr = x + (rbase0 + g * 16 + m16) * H_;
        v2f afrag[8];
#pragma unroll
        for (int kk = 0; kk < 8; ++kk)
            afrag[kk] = *(const v2f*)&xr[kk * 4 + khalf];

#pragma unroll
        for (int jt = 0; jt < 2; ++jt) {
            v8f acc = {0.f, 0.f, 0.f, 0.f, 0.f, 0.f, 0.f, 0.f};
#pragma unroll
            for (int kk = 0; kk < 8; ++kk)
                acc = __builtin_amdgcn_wmma_f32_16x16x4_f32(
                    false, afrag[kk], false, bfrag[jt][kk], (short)0, acc, false, false);
#pragma unroll
            for (int v = 0; v < 8; ++v) {
                int m = ((lane >> 4) << 3) + v;
                lds[wv][m][jt * 16 + m16] = fast_tanh(acc[v] + bias[jt]);
            }
        }
        __asm__ volatile("" ::: "memory");

        float* orow = out + (rbase0 + g * 16) * H_ + lane;
#pragma unroll 4
        for (int t = 0; t < 16; ++t) {
            float u = lds[wv][t][lane];
            float y = Dh * u;
#pragma unroll
            for (int n = 0; n < N_; ++n) {
                float r = fmaf(lam_re[n], sre[n], fmaf(-lam_im[n], sim[n], u));
                float i = fmaf(lam_re[n], sim[n], lam_im[n] * sre[n]);
                sre[n] = r; sim[n] = i;
                y = fmaf(Ctre[n], r, fmaf(-Ctim[n], i, y));
            }
            orow[t * H_] = fast_tanh(fmaf(gg, y, bb));
        }
        __asm__ volatile("" ::: "memory");
    }
}

// ---------------------------------------------------------------------------
extern "C" void kernel_launch(void* const* d_in, const int* in_sizes, int n_in,
                              void* d_out, int out_size, void* d_ws, size_t ws_size,
                              hipStream_t stream) {
    (void)in_sizes; (void)n_in; (void)out_size; (void)ws_size;
    const float* x          = (const float*)d_in[0];
    const float* cond       = (const float*)d_in[1];
    const float* lin_w      = (const float*)d_in[2];
    const float* lin_b      = (const float*)d_in[3];
    const float* log_dt     = (const float*)d_in[4];
    const float* log_A_real = (const float*)d_in[5];
    const float* A_imag     = (const float*)d_in[6];
    const float* C_re       = (const float*)d_in[7];
    const float* C_im       = (const float*)d_in[8];
    const float* Dvec       = (const float*)d_in[9];
    const float* film_w     = (const float*)d_in[10];
    const float* film_b     = (const float*)d_in[11];
    float* out   = (float*)d_out;
    float* carry = (float*)d_ws;   // needs B*CHUNKS*H*8 floats = 4 MB

    const int nwaves = B_ * CHUNKS_;                  // 4096
    dim3 grid(nwaves / WAVES_PER_BLOCK);              // 512 blocks x 256 threads

    s4d_phase1<<<grid, BLOCK_THREADS, 0, stream>>>(x, lin_w, lin_b, log_dt,
                                                   log_A_real, A_imag, carry);
    s4d_phase2<<<1, B_ * H_, 0, stream>>>(log_dt, log_A_real, A_imag, carry);
    s4d_phase3<<<grid, BLOCK_THREADS, 0, stream>>>(x, lin_w, lin_b, log_dt,
                                                   log_A_real, A_imag, C_re, C_im,
                                                   Dvec, cond, film_w, film_b,
                                                   carry, out);
}